// GraphAutoencoder_65704409694294
// MI455X (gfx1250) — compile-verified
//
#include <hip/hip_runtime.h>

typedef __attribute__((ext_vector_type(2))) float v2f;
typedef __attribute__((ext_vector_type(8))) float v8f;

// ---------------- norm precompute ----------------

__global__ void k_fill1(float* __restrict__ p, int n) {
  int i = blockIdx.x * blockDim.x + threadIdx.x;
  if (i < n) p[i] = 1.0f;  // self-loop contributes 1 to every degree
}

__global__ void k_deg_accum(const int* __restrict__ dst, float* __restrict__ deg, int e) {
  int i = blockIdx.x * blockDim.x + threadIdx.x;
  if (i < e) atomicAdd(&deg[dst[i]], 1.0f);
}

__global__ void k_rsqrt(float* __restrict__ p, int n) {
  int i = blockIdx.x * blockDim.x + threadIdx.x;
  if (i < n) p[i] = rsqrtf(p[i]);  // deg >= 1 always (self-loop)
}

__global__ void k_edge_norm(const int* __restrict__ src, const int* __restrict__ dst,
                            const float* __restrict__ dis, float* __restrict__ nrm, int e) {
  int i = blockIdx.x * blockDim.x + threadIdx.x;
  if (i < e) nrm[i] = dis[src[i]] * dis[dst[i]];
}

// ---- fp32 WMMA GEMM: async-staged, column-major (transposed) LDS W tile ----
// Block: 256 threads = 8 waves; one 32-wide column group (blockIdx.y), 8 row tiles.
// LDS layout: tileW[col][k] with row stride K+2 floats (even -> 8B aligned b64 loads;
// odd-ish bank spread -> conflict-free). B fragments become single ds_load_b64 each.
// Outputs: T = act(A)@W and Hout = bias + dis^2 * T (fused self-loop init).

template <int K, bool RELU>
__global__ void k_gemm_wmma(const float* __restrict__ A, const float* __restrict__ W,
                            const float* __restrict__ dis, const float* __restrict__ bias,
                            float* __restrict__ T, float* __restrict__ Hout,
                            int Mtiles, int Nw) {
  constexpr int LDK = K + 2;            // padded column stride (floats)
  __shared__ float tileW[32 * LDK];

  const int t  = threadIdx.x;
  const int ng = blockIdx.y;

  // Async transposing stage: W[k][ng*32+c] -> tileW[c*LDK + k].
  // Consecutive lanes read consecutive global columns (coalesced b32 streams).
  constexpr int PASSES = K / 8;         // K*32 elements / 256 threads
#pragma unroll
  for (int p = 0; p < PASSES; ++p) {
    int f = p * 256 + t;                // flat element index
    int k = f >> 5;
    int c = f & 31;
    unsigned lds_addr = (unsigned)(uintptr_t)(&tileW[c * LDK + k]);
    const float* gp = W + (size_t)k * Nw + (ng << 5) + c;
    asm volatile("global_load_async_to_lds_b32 %0, %1, off"
                 :: "v"(lds_addr), "v"(gp) : "memory");
  }
  asm volatile("s_wait_asynccnt 0x0" ::: "memory");
  __syncthreads();

  const int lane = t & 31;
  const int wave = t >> 5;
  const int mt   = blockIdx.x * 8 + wave;
  if (mt >= Mtiles) return;             // wave-uniform: EXEC all-1s at every WMMA

  const int half = lane >> 4;
  const int l16  = lane & 15;
  const float* arow = A + (size_t)(mt * 16 + l16) * K + 2 * half;
  const float* b0p  = &tileW[(l16)      * LDK + 2 * half];
  const float* b1p  = &tileW[(l16 + 16) * LDK + 2 * half];

  v8f c0 = {};
  v8f c1 = {};
#pragma unroll 8
  for (int k0 = 0; k0 < K; k0 += 4) {
    float ax = arow[k0];
    float ay = arow[k0 + 1];
    if (RELU) { ax = fmaxf(ax, 0.0f); ay = fmaxf(ay, 0.0f); }
    v2f a; a.x = ax; a.y = ay;
    v2f b0 = *(const v2f*)(b0p + k0);   // {K=k0+2h, K=k0+2h+1} : one ds_load_b64
    v2f b1 = *(const v2f*)(b1p + k0);
    c0 = __builtin_amdgcn_wmma_f32_16x16x4_f32(false, a, false, b0, (short)0, c0, false, false);
    c1 = __builtin_amdgcn_wmma_f32_16x16x4_f32(false, a, false, b1, (short)0, c1, false, false);
  }

  // Fused epilogue: write T and the self-loop/bias-initialized accumulator.
  const int colbase = (ng << 5) + l16;
  const float bia0 = bias[colbase];
  const float bia1 = bias[colbase + 16];
  const size_t rowbase = (size_t)(mt * 16 + half * 8);
  float* tp = T    + rowbase * Nw + colbase;
  float* hp = Hout + rowbase * Nw + colbase;
#pragma unroll
  for (int v = 0; v < 8; ++v) {
    float d  = dis[rowbase + v];
    float d2 = d * d;
    tp[(size_t)v * Nw]      = c0[v];
    tp[(size_t)v * Nw + 16] = c1[v];
    hp[(size_t)v * Nw]      = bia0 + d2 * c0[v];
    hp[(size_t)v * Nw + 16] = bia1 + d2 * c1[v];
  }
}

// ---------------- edge scatter: out[dst,:] += norm * T[src,:] ----------------
// float4 gather + 4 L2-side f32 atomics per thread; blockDim.x = C/4 channels.

__global__ void k_scatter4(const int* __restrict__ src, const int* __restrict__ dst,
                           const float* __restrict__ nrm, const float4* __restrict__ t4,
                           float* __restrict__ out, int e, int cq) {
  int ch   = threadIdx.x;                    // channel quad index, 0..cq-1
  int edge = blockIdx.x * blockDim.y + threadIdx.y;
  if (edge >= e) return;
  int s = src[edge];
  int d = dst[edge];
  float nv = nrm[edge];
  float4 v = t4[(size_t)s * cq + ch];
  float* o = out + ((size_t)d * cq + ch) * 4;
  atomicAdd(o + 0, nv * v.x);
  atomicAdd(o + 1, nv * v.y);
  atomicAdd(o + 2, nv * v.z);
  atomicAdd(o + 3, nv * v.w);
}

// ---------------- host orchestration ----------------

extern "C" void kernel_launch(void* const* d_in, const int* in_sizes, int n_in,
                              void* d_out, int out_size, void* d_ws, size_t ws_size,
                              hipStream_t stream) {
  (void)n_in; (void)out_size; (void)ws_size;
  const int D[7] = {256, 128, 64, 32, 64, 128, 256};

  const float* x  = (const float*)d_in[0];
  const int*   ei = (const int*)d_in[1];
  const int N = in_sizes[0] / D[0];
  const int E = in_sizes[1] / 2;
  const int Mtiles = N / 16;
  const int* src  = ei;        // edge_index[0]
  const int* dstp = ei + E;    // edge_index[1]

  const float* Wl[6];
  const float* bl[6];
  for (int i = 0; i < 6; ++i) {
    Wl[i] = (const float*)d_in[2 + 2 * i];
    bl[i] = (const float*)d_in[3 + 2 * i];
  }

  // workspace: dis(N), norm(E), T(N x 256), two ping-pong h buffers(N x 128)
  char*  ws  = (char*)d_ws;
  size_t off = 0;
  auto alloc = [&](size_t bytes) -> float* {
    float* p = (float*)(ws + off);
    off = (off + bytes + 255) & ~(size_t)255;
    return p;
  };
  float* dis = alloc((size_t)N * 4);
  float* nrm = alloc((size_t)E * 4);
  float* T   = alloc((size_t)N * 256 * 4);
  float* H1  = alloc((size_t)N * 128 * 4);
  float* H2  = alloc((size_t)N * 128 * 4);

  const int TB = 256;
  k_fill1    <<<(N + TB - 1) / TB, TB, 0, stream>>>(dis, N);
  k_deg_accum<<<(E + TB - 1) / TB, TB, 0, stream>>>(dstp, dis, E);
  k_rsqrt    <<<(N + TB - 1) / TB, TB, 0, stream>>>(dis, N);
  k_edge_norm<<<(E + TB - 1) / TB, TB, 0, stream>>>(src, dstp, dis, nrm, E);

  const float* hin = x;
  float* houts[6] = {H1, H2, H1, H2, H1, (float*)d_out};
  for (int l = 0; l < 6; ++l) {
    const int C = D[l + 1];
    float* hout = houts[l];

    // T = act(hin) @ W ; hout = bias + dis^2 * T   (relu folded into A-read)
    dim3 gg((Mtiles + 7) / 8, C / 32);
    switch (l) {
      case 0: k_gemm_wmma<256, false><<<gg, 256, 0, stream>>>(hin, Wl[l], dis, bl[l], T, hout, Mtiles, C); break;
      case 1: k_gemm_wmma<128, true ><<<gg, 256, 0, stream>>>(hin, Wl[l], dis, bl[l], T, hout, Mtiles, C); break;
      case 2: k_gemm_wmma< 64, true ><<<gg, 256, 0, stream>>>(hin, Wl[l], dis, bl[l], T, hout, Mtiles, C); break;
      case 3: k_gemm_wmma< 32, false><<<gg, 256, 0, stream>>>(hin, Wl[l], dis, bl[l], T, hout, Mtiles, C); break;
      case 4: k_gemm_wmma< 64, true ><<<gg, 256, 0, stream>>>(hin, Wl[l], dis, bl[l], T, hout, Mtiles, C); break;
      default:k_gemm_wmma<128, true ><<<gg, 256, 0, stream>>>(hin, Wl[l], dis, bl[l], T, hout, Mtiles, C); break;
    }

    // edge scatter-add into hout (pre-activation; next layer applies relu on read)
    int  cq  = C / 4;
    dim3 sb(cq, 256 / cq);
    int  epb = 256 / cq;
    k_scatter4<<<(E + epb - 1) / epb, sb, 0, stream>>>(src, dstp, nrm, (const float4*)T, hout, E, cq);

    hin = hout;
  }
}